// StateSpaceAttentionV2_44358422233466
// MI455X (gfx1250) — compile-verified
//
#include <hip/hip_runtime.h>
#include <math.h>

#define DIM   768
#define TSEQ  4096
#define NREG  4088
#define NSS   8
#define INNER 192
#define SSD   1536   // NSS*INNER

typedef __attribute__((ext_vector_type(2))) float v2f;
typedef __attribute__((ext_vector_type(8))) float v8f;
typedef __attribute__((ext_vector_type(4))) int   v4i;

// ---------------------------------------------------------------------------
// CDNA5 async memory -> LDS copy (ASYNCcnt-tracked), builtin confirmed on
// this toolchain: (v4i AS1*, v4i AS3*, const int offset, const int cpol).
// ---------------------------------------------------------------------------
__device__ __forceinline__ void async_g2l_b128(const float* g, float* l) {
#if __has_builtin(__builtin_amdgcn_global_load_async_to_lds_b128)
  __builtin_amdgcn_global_load_async_to_lds_b128(
      (__attribute__((address_space(1))) v4i*)(g),
      (__attribute__((address_space(3))) v4i*)(l), 0, 0);
#else
  unsigned lo = (unsigned)(size_t)(__attribute__((address_space(3))) char*)l;
  asm volatile("global_load_async_to_lds_b128 %0, %1, off"
               :: "v"(lo), "v"(g) : "memory");
#endif
}

#if __has_builtin(__builtin_amdgcn_s_wait_asynccnt)
#define WAIT_ASYNC(n) __builtin_amdgcn_s_wait_asynccnt(n)
#else
#define WAIT_ASYNC(n) asm volatile("s_wait_asynccnt %0" :: "i"(n) : "memory")
#endif

// ---------------------------------------------------------------------------
// Kernel A: partial column sums of reg rows (for mean(reg)@Wk trick)
// ---------------------------------------------------------------------------
__global__ __launch_bounds__(256) void k_colsum(const float* __restrict__ x,
                                                float* __restrict__ msum) {
  int b  = blockIdx.y;
  int r0 = blockIdx.x * 128;
  int r1 = r0 + 128; if (r1 > NREG) r1 = NREG;
  int tid = threadIdx.x;
  float s0 = 0.f, s1 = 0.f, s2 = 0.f;
  const float* xb = x + ((size_t)b * TSEQ + r0) * DIM;
  for (int r = 0; r < r1 - r0; ++r) {
    const float* row = xb + (size_t)r * DIM;
    s0 += row[tid];
    s1 += row[tid + 256];
    s2 += row[tid + 512];
  }
  atomicAdd(&msum[b * DIM + tid],       s0);
  atomicAdd(&msum[b * DIM + tid + 256], s1);
  atomicAdd(&msum[b * DIM + tid + 512], s2);
}

// ---------------------------------------------------------------------------
// Kernel B1: ssu[b] = mean(reg_b) @ Wk + bk   (16 x 1536)
// ---------------------------------------------------------------------------
__global__ __launch_bounds__(256) void k_ssu(const float* __restrict__ msum,
                                             const float* __restrict__ Wk,
                                             const float* __restrict__ bk,
                                             float* __restrict__ ssu) {
  int b = blockIdx.x, tid = threadIdx.x;
  __shared__ float mean[DIM];
  for (int d = tid; d < DIM; d += 256) mean[d] = msum[b * DIM + d] * (1.0f / NREG);
  __syncthreads();
  for (int j = tid; j < SSD; j += 256) {
    float acc = bk[j];
    for (int d = 0; d < DIM; ++d) acc += mean[d] * Wk[(size_t)d * SSD + j];
    ssu[b * SSD + j] = acc;
  }
}

// ---------------------------------------------------------------------------
// Kernel B2: new_ss[b,s,:] = x[b,4088+s,:] + ssu[b,s,:] @ Wss_out + bss_out
// also writes output tail rows 4088..4095
// ---------------------------------------------------------------------------
__global__ __launch_bounds__(256) void k_newss(const float* __restrict__ x,
                                               const float* __restrict__ ssu,
                                               const float* __restrict__ Wss,
                                               const float* __restrict__ bss,
                                               float* __restrict__ nss,
                                               float* __restrict__ out) {
  int b = blockIdx.x, tid = threadIdx.x;
  __shared__ float u[SSD];
  for (int j = tid; j < SSD; j += 256) u[j] = ssu[b * SSD + j];
  __syncthreads();
  for (int s = 0; s < NSS; ++s) {
    const float* us = u + s * INNER;
    for (int d = tid; d < DIM; d += 256) {
      float acc = x[((size_t)b * TSEQ + NREG + s) * DIM + d] + bss[d];
      for (int k = 0; k < INNER; ++k) acc += us[k] * Wss[(size_t)k * DIM + d];
      nss[((size_t)b * NSS + s) * DIM + d] = acc;
      out[((size_t)b * TSEQ + NREG + s) * DIM + d] = acc;
    }
  }
}

// ---------------------------------------------------------------------------
// Kernel B3: Qt[b,j,s] = sum_d Wq[j,d]*new_ss[b,s,d]  and  cb[b,s] = bq . ss_s
// ---------------------------------------------------------------------------
__global__ __launch_bounds__(256) void k_qt(const float* __restrict__ Wq,
                                            const float* __restrict__ bq,
                                            const float* __restrict__ nss,
                                            float* __restrict__ Qt,
                                            float* __restrict__ cb) {
  int b = blockIdx.x, tid = threadIdx.x;
  __shared__ float ss[NSS * DIM];
  for (int i = tid; i < NSS * DIM; i += 256) ss[i] = nss[(size_t)b * NSS * DIM + i];
  __syncthreads();
  for (int idx = tid; idx < DIM * NSS; idx += 256) {
    int j = idx >> 3, s = idx & 7;
    float acc = 0.f;
    const float* wr = Wq + (size_t)j * DIM;
    const float* sr = ss + s * DIM;
    for (int d = 0; d < DIM; ++d) acc += wr[d] * sr[d];
    Qt[(size_t)b * DIM * NSS + idx] = acc;
  }
  if (tid < NSS) {
    float acc = 0.f;
    for (int d = 0; d < DIM; ++d) acc += bq[d] * ss[tid * DIM + d];
    cb[b * NSS + tid] = acc;
  }
}

// ---------------------------------------------------------------------------
// Kernel C: fused  x_v = reg@Wv + bv ;  sim -> softmax -> upd ; relu ; store
// grid (128 m-tiles, 16 batches), 256 threads = 8 waves.
// M_TILE=32: waves = {mi in 0..1} x {ni in 0..3}; wave C slab = 16 x 192.
// Wv streamed via DOUBLE-BUFFERED GLOBAL_LOAD_ASYNC_TO_LDS_B128: chunk c+1's
// copies are in flight (ASYNCcnt) while chunk c feeds the WMMAs; in-order
// async completion lets "s_wait_asynccnt <= KC" retire chunk c exactly.
// LDS row stride 776 (776 % 64 == 8 -> lane-halves hit disjoint banks).
// WMMA: V_WMMA_F32_16X16X4_F32; all 13 B-fragments of a k-step are loaded
// into VGPRs before the 13 WMMAs so LDS waits batch instead of serializing.
// ---------------------------------------------------------------------------
#define MTILE 32
#define KC    8
#define KP    776
#define NCH   (DIM / KC)   // 96 chunks

__global__ __launch_bounds__(256) void k_main(const float* __restrict__ x,
                                              const float* __restrict__ Wv,
                                              const float* __restrict__ bv,
                                              const float* __restrict__ nss,
                                              const float* __restrict__ Qt,
                                              const float* __restrict__ cb,
                                              float* __restrict__ out) {
  __shared__ float Bs[2][KC * KP];         // 2*8*776*4 = 49664 B
  __shared__ float simS[MTILE * NSS];
  __shared__ float attnS[MTILE * NSS];

  int b   = blockIdx.y;
  int m0  = blockIdx.x * MTILE;
  int tid = threadIdx.x;
  int wave = tid >> 5, lane = tid & 31;
  int mi = wave & 1, ni = wave >> 1;
  int h  = lane >> 4, l16 = lane & 15;
  int nb = ni * 192;

  int rowA = m0 + mi * 16 + l16;
  if (rowA >= NREG) rowA = NREG - 1;       // clamp; invalid rows not stored
  const float* xA  = x  + ((size_t)b * TSEQ + rowA) * DIM;
  const float* QtB = Qt + (size_t)b * DIM * NSS;

  v8f acc[12];
#pragma unroll
  for (int t = 0; t < 12; ++t) acc[t] = {};
  v8f sacc = {};

  // prologue: stage chunk 0 into buffer 0
  if (tid < 192) {                          // wave-uniform (waves 0..5)
#pragma unroll
    for (int k = 0; k < KC; ++k)
      async_g2l_b128(Wv + (size_t)k * DIM + tid * 4, &Bs[0][k * KP + tid * 4]);
  }

  for (int c = 0; c < NCH; ++c) {
    int kk = c * KC;
    const float* cur = Bs[c & 1];
    if (c + 1 < NCH) {
      // issue chunk c+1 into the other buffer, then retire chunk c
      if (tid < 192) {
#pragma unroll
        for (int k = 0; k < KC; ++k)
          async_g2l_b128(Wv + (size_t)(kk + KC + k) * DIM + tid * 4,
                         &Bs[(c + 1) & 1][k * KP + tid * 4]);
      }
      WAIT_ASYNC(KC);                       // chunk c complete (in-order)
    } else {
      WAIT_ASYNC(0);
    }
    __syncthreads();

#pragma unroll
    for (int k4 = 0; k4 < KC; k4 += 4) {
      int kb = k4 + 2 * h;
      v2f af = *(const v2f*)(xA + kk + kb);          // A frag: one b64/lane
      v2f qf;                                         // sim B frag (Qt cols)
      qf.x = (l16 < NSS) ? QtB[(size_t)(kk + kb)     * NSS + l16] : 0.0f;
      qf.y = (l16 < NSS) ? QtB[(size_t)(kk + kb + 1) * NSS + l16] : 0.0f;
      v2f bf[12];                                     // batch all LDS loads
#pragma unroll
      for (int t = 0; t < 12; ++t) {
        bf[t].x = cur[kb       * KP + nb + t * 16 + l16];
        bf[t].y = cur[(kb + 1) * KP + nb + t * 16 + l16];
      }
      sacc = __builtin_amdgcn_wmma_f32_16x16x4_f32(
          false, af, false, qf, (short)0, sacc, false, false);
#pragma unroll
      for (int t = 0; t < 12; ++t)
        acc[t] = __builtin_amdgcn_wmma_f32_16x16x4_f32(
            false, af, false, bf[t], (short)0, acc[t], false, false);
    }
    __syncthreads();
  }

  // sim tile -> LDS with bias and 1/sqrt(768) scale (only ni==0 waves write)
  const float scale = 0.03608439182435161f;
  if (ni == 0 && l16 < NSS) {
    float cbs = cb[b * NSS + l16];
#pragma unroll
    for (int r = 0; r < 8; ++r)
      simS[(mi * 16 + r + 8 * h) * NSS + l16] = (sacc[r] + cbs) * scale;
  }
  __syncthreads();

  // softmax over 8 slots, one row per thread
  if (tid < MTILE) {
    float v[NSS], mx = -1e30f;
#pragma unroll
    for (int s = 0; s < NSS; ++s) { v[s] = simS[tid * NSS + s]; mx = fmaxf(mx, v[s]); }
    float sum = 0.f;
#pragma unroll
    for (int s = 0; s < NSS; ++s) { v[s] = __expf(v[s] - mx); sum += v[s]; }
    float inv = 1.0f / sum;
#pragma unroll
    for (int s = 0; s < NSS; ++s) attnS[tid * NSS + s] = v[s] * inv;
  }
  __syncthreads();

  // reuse Bs LDS for new_ss[b] (8 x 768 = 6144 floats fits in both buffers)
  float* nsS = &Bs[0][0];
  for (int i = tid; i < NSS * DIM; i += 256) nsS[i] = nss[(size_t)b * NSS * DIM + i];
  __syncthreads();

  // epilogue: out = relu(acc + bv + attn @ new_ss)
#pragma unroll
  for (int t = 0; t < 12; ++t) {
    int n = nb + t * 16 + l16;
    float bvn = bv[n];
#pragma unroll
    for (int r = 0; r < 8; ++r) {
      int ml  = r + 8 * h;
      int row = m0 + mi * 16 + ml;
      if (row < NREG) {
        float upd = 0.f;
#pragma unroll
        for (int s = 0; s < NSS; ++s)
          upd += attnS[(mi * 16 + ml) * NSS + s] * nsS[s * DIM + n];
        float val = acc[t][r] + bvn + upd;
        out[((size_t)b * TSEQ + row) * DIM + n] = fmaxf(val, 0.0f);
      }
    }
  }
}

// ---------------------------------------------------------------------------
extern "C" void kernel_launch(void* const* d_in, const int* in_sizes, int n_in,
                              void* d_out, int out_size, void* d_ws, size_t ws_size,
                              hipStream_t stream) {
  const float* x   = (const float*)d_in[0];
  const float* Wq  = (const float*)d_in[1];
  const float* bq  = (const float*)d_in[2];
  const float* Wk  = (const float*)d_in[3];
  const float* bk  = (const float*)d_in[4];
  const float* Wv  = (const float*)d_in[5];
  const float* bv  = (const float*)d_in[6];
  const float* Wss = (const float*)d_in[7];
  const float* bss = (const float*)d_in[8];
  float* out = (float*)d_out;

  float* ws   = (float*)d_ws;
  float* msum = ws;                       // 16*768
  float* ssu  = msum + 16 * DIM;          // 16*1536
  float* nss  = ssu  + 16 * SSD;          // 16*8*768
  float* Qt   = nss  + 16 * NSS * DIM;    // 16*768*8
  float* cb   = Qt   + 16 * DIM * NSS;    // 16*8

  (void)hipMemsetAsync(msum, 0, 16 * DIM * sizeof(float), stream);
  k_colsum<<<dim3(32, 16), 256, 0, stream>>>(x, msum);
  k_ssu  <<<16, 256, 0, stream>>>(msum, Wk, bk, ssu);
  k_newss<<<16, 256, 0, stream>>>(x, ssu, Wss, bss, nss, out);
  k_qt   <<<16, 256, 0, stream>>>(Wq, bq, nss, Qt, cb);
  k_main <<<dim3(128, 16), 256, 0, stream>>>(x, Wv, bv, nss, Qt, cb, out);
}